// Scalable_PacketTimeLSTM_3_31190052504106
// MI455X (gfx1250) — compile-verified
//
#include <hip/hip_runtime.h>
#include <math.h>

typedef __attribute__((ext_vector_type(2))) float v2f;
typedef __attribute__((ext_vector_type(8))) float v8f;

#define HDIM 64          // H
#define G3   192         // 3H
#define KW   65          // H+1 (matvec K)
#define BOUNDARY (-0.01f)

// Fast transcendentals on the hardware v_exp path.
__device__ __forceinline__ float sigmoidf(float v) {
    return 1.0f / (1.0f + __expf(-v));
}
__device__ __forceinline__ float tanh_fast(float v) {
    // tanh(x) = 1 - 2/(exp(2x)+1); saturates correctly at +/-inf.
    return 1.0f - 2.0f / (__expf(2.0f * v) + 1.0f);
}

// ---------------------------------------------------------------------------
// Kernel 0: initialize outputs. H_curr <- 0, c_global_new <- c_global.
// ---------------------------------------------------------------------------
__global__ void init_out_kernel(const float* __restrict__ c_global,
                                float* __restrict__ Hout,
                                float* __restrict__ CGout,
                                int total) {
    int i = blockIdx.x * blockDim.x + threadIdx.x;
    if (i < total) {
        Hout[i]  = 0.0f;
        CGout[i] = c_global[i];
    }
}

// ---------------------------------------------------------------------------
// Kernel 1: per-feature LSTM cell. One wave (32 lanes) per feature.
// Matvec cur_out[192] = W[192x65] * [x; h_prev] via V_WMMA_F32_16X16X4_F32
// with broadcast-B (all 16 N columns identical). Branchless tail chunk.
// ---------------------------------------------------------------------------
__global__ __launch_bounds__(256) void lstm_main_kernel(
    const float* __restrict__ X, const float* __restrict__ delta,
    const float* __restrict__ Ht, const float* __restrict__ c_t,
    const float* __restrict__ Wg,   // lstm_weights  [F,192,65]
    const float* __restrict__ Bg,   // lstm_bias     [F,192]
    const float* __restrict__ WxT,  // lstm_xT_weights [F,128]
    const float* __restrict__ BxT,  // lstm_xT_bias  [F,128]
    const float* __restrict__ WdT,  // lstm_delT_weights [F,192]
    const float* __restrict__ Wci,  // [F,64]
    const float* __restrict__ Wco,  // [F,64]
    const int* __restrict__ feat,
    float* __restrict__ Hout, float* __restrict__ CGout,
    int n) {
    __shared__ float s_inp[8][72];   // [x, h(64), zeros(..71)]; rows 8B-aligned
    __shared__ float s_cur[8][192];  // cur_out per wave

    const int lane = threadIdx.x & 31;
    const int w    = threadIdx.x >> 5;
    const int nIdx = blockIdx.x * 8 + w;
    if (nIdx >= n) return;                 // wave-uniform exit

    const int f    = feat[nIdx];
    const float x  = X[f];
    const float dl = delta[f];

    // Stage cur_input = [x, h_prev, 0-pad] into LDS (read only by this wave).
    s_inp[w][1 + lane]      = Ht[(size_t)f * HDIM + lane];
    s_inp[w][1 + 32 + lane] = Ht[(size_t)f * HDIM + 32 + lane];
    if (lane == 0) s_inp[w][0] = x;
    if (lane < 7)  s_inp[w][65 + lane] = 0.0f;

    const int   m     = lane & 15;   // M row within tile
    const int   hi    = lane >> 4;   // K sub-pair selector (K = 4c + 2*hi + {0,1})
    const float hisel = (hi == 0) ? 1.0f : 0.0f;   // tail-chunk mask (branchless)
    const float* __restrict__ Wf = Wg + (size_t)f * G3 * KW;

    // Hoist the 17 broadcast-B chunks into registers (independent of tile t).
    v2f Bv[17];
    #pragma unroll
    for (int c = 0; c < 16; ++c) {
        const int k = 4 * c + 2 * hi;
        Bv[c][0] = s_inp[w][k];
        Bv[c][1] = s_inp[w][k + 1];
    }
    Bv[16][0] = s_inp[w][64] * hisel;   // k=64 valid only for the hi==0 group
    Bv[16][1] = 0.0f;

    for (int t = 0; t < 12; ++t) {                 // 12 tiles of 16 gate rows
        v8f acc = {};
        const float* __restrict__ rowp = Wf + (size_t)(t * 16 + m) * KW;
        #pragma unroll
        for (int c = 0; c < 16; ++c) {             // K chunks 0..63
            const int k = 4 * c + 2 * hi;
            v2f A; A[0] = rowp[k]; A[1] = rowp[k + 1];
            acc = __builtin_amdgcn_wmma_f32_16x16x4_f32(
                false, A, false, Bv[c], (short)0, acc, false, false);
        }
        {   // tail chunk k==64: rowp[64] is in-bounds for every lane's row;
            // zero the hi-group contribution by value, no EXEC branching.
            v2f A; A[0] = rowp[64] * hisel; A[1] = 0.0f;
            acc = __builtin_amdgcn_wmma_f32_16x16x4_f32(
                false, A, false, Bv[16], (short)0, acc, false, false);
        }
        // All 16 N columns identical: lane 0 holds D[M=0..7], lane 16 D[M=8..15].
        if (m == 0) {
            const int base = t * 16 + hi * 8;
            #pragma unroll
            for (int r = 0; r < 8; ++r) s_cur[w][base + r] = acc[r];
        }
    }

    // Elementwise tail: 2 channels per lane.
    #pragma unroll
    for (int p = 0; p < 2; ++p) {
        const int j = lane + p * 32;
        const float ct = c_t[j];

        float gi = s_cur[w][j] + Bg[(size_t)f * G3 + j]
                 + Wci[(size_t)f * HDIM + j] * ct;
        gi = sigmoidf(gi);
        const float gpre = tanh_fast(s_cur[w][128 + j] + Bg[(size_t)f * G3 + 128 + j]);

        const float xm1 = WxT[(size_t)f * 128 + j]      * x + BxT[(size_t)f * 128 + j];
        const float xm2 = WxT[(size_t)f * 128 + 64 + j] * x + BxT[(size_t)f * 128 + 64 + j];

        const float w1 = WdT[(size_t)f * G3 + j];
        float       w2 = WdT[(size_t)f * G3 + 64 + j];
        if (j == 0) w2 = fminf(w2, BOUNDARY);           // clip row H of delT weights
        const float wo = WdT[(size_t)f * G3 + 128 + j];

        const float T1 = sigmoidf(xm1 + sigmoidf(w1 * dl));
        const float T2 = sigmoidf(xm2 + sigmoidf(w2 * dl));

        const float ctil = (1.0f - gi * T1) * ct + gi * T1 * gpre;
        const float cnew = (1.0f - gi)      * ct + gi * T2 * gpre;

        const float go = sigmoidf(s_cur[w][64 + j] + Bg[(size_t)f * G3 + 64 + j]
                                  + Wco[(size_t)f * HDIM + j] * ctil + wo * dl);
        const float hnew = go * tanh_fast(ctil);

        Hout [(size_t)f * HDIM + j] = hnew;
        CGout[(size_t)f * HDIM + j] = cnew;
    }
}

// ---------------------------------------------------------------------------
// Kernel 2: C_curr[j] = mean over features of c_new (read back from CGout).
// Deterministic tree reduction, one block per channel.
// ---------------------------------------------------------------------------
__global__ void mean_kernel(const float* __restrict__ CGout,
                            const int* __restrict__ feat,
                            float* __restrict__ Cmean, int n) {
    __shared__ float red[256];
    const int j = blockIdx.x;
    float s = 0.0f;
    for (int i = threadIdx.x; i < n; i += 256)
        s += CGout[(size_t)feat[i] * HDIM + j];
    red[threadIdx.x] = s;
    __syncthreads();
    for (int off = 128; off > 0; off >>= 1) {
        if (threadIdx.x < off) red[threadIdx.x] += red[threadIdx.x + off];
        __syncthreads();
    }
    if (threadIdx.x == 0) Cmean[j] = red[0] / (float)n;
}

extern "C" void kernel_launch(void* const* d_in, const int* in_sizes, int n_in,
                              void* d_out, int out_size, void* d_ws, size_t ws_size,
                              hipStream_t stream) {
    const float* X     = (const float*)d_in[0];
    const float* delta = (const float*)d_in[1];
    const float* Ht    = (const float*)d_in[2];
    const float* c_t   = (const float*)d_in[3];
    const float* cg    = (const float*)d_in[4];
    const float* Wg    = (const float*)d_in[5];
    const float* Bg    = (const float*)d_in[6];
    const float* WxT   = (const float*)d_in[7];
    const float* BxT   = (const float*)d_in[8];
    const float* WdT   = (const float*)d_in[9];
    const float* Wci   = (const float*)d_in[10];
    const float* Wco   = (const float*)d_in[11];
    const int*   feat  = (const int*)d_in[12];

    const int F = in_sizes[0];          // 16384
    const int n = in_sizes[12];         // 8192

    float* Hout  = (float*)d_out;
    float* CGout = Hout + (size_t)F * HDIM;
    float* Cmean = CGout + (size_t)F * HDIM;

    const int total = F * HDIM;
    init_out_kernel<<<(total + 255) / 256, 256, 0, stream>>>(cg, Hout, CGout, total);

    lstm_main_kernel<<<(n + 7) / 8, 256, 0, stream>>>(
        X, delta, Ht, c_t, Wg, Bg, WxT, BxT, WdT, Wci, Wco, feat,
        Hout, CGout, n);

    mean_kernel<<<HDIM, 256, 0, stream>>>(CGout, feat, Cmean, n);
}